// HeteroGraphSAGE_57750130262288
// MI455X (gfx1250) — compile-verified
//
#include <hip/hip_runtime.h>
#include <hip/hip_bf16.h>

#define NUSER 100000
#define NPC   20000
#define NURL  50000
#define EUP   250000
#define EUU   500000
#define HD    128

typedef float v2f __attribute__((ext_vector_type(2)));
typedef float v8f __attribute__((ext_vector_type(8)));

// ---------------------------------------------------------------------------
// Skinny input projection: out[n, c] = sum_f x[n,f] * W[f,c] + b[c]
// ---------------------------------------------------------------------------
__global__ void hgs_proj_kernel(const float* __restrict__ x,
                                const float* __restrict__ W,
                                const float* __restrict__ b,
                                float* __restrict__ out,
                                int N, int F, int out_ld) {
    int tid = blockIdx.x * blockDim.x + threadIdx.x;
    if (tid >= N * HD) return;
    int n = tid >> 7;
    int c = tid & (HD - 1);
    float acc = b[c];
    for (int f = 0; f < F; ++f)
        acc += x[(size_t)n * F + f] * W[f * HD + c];
    out[(size_t)n * out_ld + c] = acc;
}

// ---------------------------------------------------------------------------
// Edge scatter-add: dst[sidx[e], :] += src[gidx[e], :]; optional degree count.
// ---------------------------------------------------------------------------
__global__ void hgs_scatter_add_kernel(const float* __restrict__ src, int src_ld,
                                       const int* __restrict__ gidx,
                                       const int* __restrict__ sidx,
                                       float* __restrict__ dst, int dst_ld,
                                       float* __restrict__ cnt, int E) {
    int tid = blockIdx.x * blockDim.x + threadIdx.x;
    if (tid >= E * 32) return;
    int e = tid >> 5;
    int c = (tid & 31) << 2;
    int g = gidx[e];
    int s = sidx[e];
    const float4 v = *(const float4*)(src + (size_t)g * src_ld + c);
    float* d = dst + (size_t)s * dst_ld + c;
    atomicAdd(d + 0, v.x);
    atomicAdd(d + 1, v.y);
    atomicAdd(d + 2, v.z);
    atomicAdd(d + 3, v.w);
    if (cnt != nullptr && (tid & 31) == 0) atomicAdd(cnt + s, 1.0f);
}

// agg[n, :] /= max(cnt[n], 1)
__global__ void hgs_mean_div_kernel(float* __restrict__ agg,
                                    const float* __restrict__ cnt, int N) {
    int tid = blockIdx.x * blockDim.x + threadIdx.x;
    if (tid >= N * HD) return;
    int n = tid >> 7;
    agg[tid] = agg[tid] / fmaxf(cnt[n], 1.0f);
}

// ---------------------------------------------------------------------------
// Fused SAGE layer (32-row blocks): Out = relu(A@Wl + X@Wr + bl (+X)), 128x128.
// 8 waves; each wave owns one 16-col tile and TWO 16-row M-tiles, so every
// B-fragment feeds two v_wmma_f32_16x16x4_f32 ops (half the VMEM per WMMA).
// ---------------------------------------------------------------------------
#define LDSP 132  // padded row stride (mod 64 = 4 -> conflict-free)

__global__ void hgs_sage_fused_kernel(const float* __restrict__ A,
                                      const float* __restrict__ X,
                                      const float* __restrict__ Wl,
                                      const float* __restrict__ Wr,
                                      const float* __restrict__ bl,
                                      float* __restrict__ Out,
                                      int N, int residual) {
    __shared__ float sA[32 * LDSP];
    __shared__ float sX[32 * LDSP];
    const int row0 = blockIdx.x * 32;
    const int tid  = threadIdx.x;

    // cooperative stage of both 32x128 panels (clamped for the ragged tail)
    for (int i = tid; i < 32 * HD; i += 256) {
        int r = i >> 7, c = i & (HD - 1);
        int rr = row0 + r; if (rr >= N) rr = N - 1;
        sA[r * LDSP + c] = A[(size_t)rr * HD + c];
        sX[r * LDSP + c] = X[(size_t)rr * HD + c];
    }
    __syncthreads();

    const int wave  = tid >> 5;
    const int lane  = tid & 31;
    const int n     = wave * 16 + (lane & 15);
    const int m     = lane & 15;
    const int khalf = (lane >> 4) << 1;  // 0 or 2

    v8f acc0, acc1;
    const float bv = bl[n];
#pragma unroll
    for (int i = 0; i < 8; ++i) { acc0[i] = bv; acc1[i] = bv; }

#pragma unroll 4
    for (int k0 = 0; k0 < HD; k0 += 4) {
        v2f b;
        b.x = Wl[(size_t)(k0 + khalf) * HD + n];
        b.y = Wl[(size_t)(k0 + khalf + 1) * HD + n];
        v2f a0, a1;
        a0.x = sA[m * LDSP + k0 + khalf];
        a0.y = sA[m * LDSP + k0 + khalf + 1];
        a1.x = sA[(16 + m) * LDSP + k0 + khalf];
        a1.y = sA[(16 + m) * LDSP + k0 + khalf + 1];
        acc0 = __builtin_amdgcn_wmma_f32_16x16x4_f32(false, a0, false, b,
                                                     (short)0, acc0, false, false);
        acc1 = __builtin_amdgcn_wmma_f32_16x16x4_f32(false, a1, false, b,
                                                     (short)0, acc1, false, false);
    }
#pragma unroll 4
    for (int k0 = 0; k0 < HD; k0 += 4) {
        v2f b;
        b.x = Wr[(size_t)(k0 + khalf) * HD + n];
        b.y = Wr[(size_t)(k0 + khalf + 1) * HD + n];
        v2f a0, a1;
        a0.x = sX[m * LDSP + k0 + khalf];
        a0.y = sX[m * LDSP + k0 + khalf + 1];
        a1.x = sX[(16 + m) * LDSP + k0 + khalf];
        a1.y = sX[(16 + m) * LDSP + k0 + khalf + 1];
        acc0 = __builtin_amdgcn_wmma_f32_16x16x4_f32(false, a0, false, b,
                                                     (short)0, acc0, false, false);
        acc1 = __builtin_amdgcn_wmma_f32_16x16x4_f32(false, a1, false, b,
                                                     (short)0, acc1, false, false);
    }

#pragma unroll
    for (int r = 0; r < 8; ++r) {
        int row = (lane < 16) ? r : (r + 8);
        float v0 = acc0[r];
        float v1 = acc1[r];
        if (residual) {
            v0 += sX[row * LDSP + n];
            v1 += sX[(16 + row) * LDSP + n];
        }
        v0 = fmaxf(v0, 0.0f);
        v1 = fmaxf(v1, 0.0f);
        if (row0 + row < N)      Out[(size_t)(row0 + row) * HD + n] = v0;
        if (row0 + 16 + row < N) Out[(size_t)(row0 + 16 + row) * HD + n] = v1;
    }
}

// ---------------------------------------------------------------------------
// Generic WMMA GEMM (32-row blocks): Out = A[32,K] @ W[K,ncols] + bias
// ncols = (blockDim.x/32)*16.  Dynamic LDS: 32*(K+4) floats.
// ---------------------------------------------------------------------------
__global__ void hgs_gemm_bias_kernel(const float* __restrict__ A, int lda, int K,
                                     const float* __restrict__ W,
                                     const float* __restrict__ bias,
                                     float* __restrict__ Out, int ldo,
                                     int N, int relu) {
    extern __shared__ float sA[];   // [32][K+4]
    const int kp    = K + 4;
    const int row0  = blockIdx.x * 32;
    const int tid   = threadIdx.x;
    const int nthr  = blockDim.x;
    const int ncols = (nthr >> 5) << 4;

    for (int r = 0; r < 32; ++r) {
        int rr = row0 + r; if (rr >= N) rr = N - 1;
        for (int c = tid; c < K; c += nthr)
            sA[r * kp + c] = A[(size_t)rr * lda + c];
    }
    __syncthreads();

    const int wave  = tid >> 5;
    const int lane  = tid & 31;
    const int n     = wave * 16 + (lane & 15);
    const int m     = lane & 15;
    const int khalf = (lane >> 4) << 1;

    v8f acc0, acc1;
    const float bv = bias[n];
#pragma unroll
    for (int i = 0; i < 8; ++i) { acc0[i] = bv; acc1[i] = bv; }

#pragma unroll 4
    for (int k0 = 0; k0 < K; k0 += 4) {
        v2f b;
        b.x = W[(size_t)(k0 + khalf) * ncols + n];
        b.y = W[(size_t)(k0 + khalf + 1) * ncols + n];
        v2f a0, a1;
        a0.x = sA[m * kp + k0 + khalf];
        a0.y = sA[m * kp + k0 + khalf + 1];
        a1.x = sA[(16 + m) * kp + k0 + khalf];
        a1.y = sA[(16 + m) * kp + k0 + khalf + 1];
        acc0 = __builtin_amdgcn_wmma_f32_16x16x4_f32(false, a0, false, b,
                                                     (short)0, acc0, false, false);
        acc1 = __builtin_amdgcn_wmma_f32_16x16x4_f32(false, a1, false, b,
                                                     (short)0, acc1, false, false);
    }

#pragma unroll
    for (int r = 0; r < 8; ++r) {
        int row = (lane < 16) ? r : (r + 8);
        float v0 = acc0[r];
        float v1 = acc1[r];
        if (relu) { v0 = fmaxf(v0, 0.0f); v1 = fmaxf(v1, 0.0f); }
        if (row0 + row < N)      Out[(size_t)(row0 + row) * ldo + n] = v0;
        if (row0 + 16 + row < N) Out[(size_t)(row0 + 16 + row) * ldo + n] = v1;
    }
}

// Final 64 -> 2 head: one thread per user row.
__global__ void hgs_head_kernel(const float* __restrict__ h,
                                const float* __restrict__ Wc2,
                                const float* __restrict__ bc2,
                                float* __restrict__ out, int N) {
    int nrow = blockIdx.x * blockDim.x + threadIdx.x;
    if (nrow >= N) return;
    float a0 = bc2[0], a1 = bc2[1];
    const float* hr = h + (size_t)nrow * 64;
#pragma unroll
    for (int k = 0; k < 64; ++k) {
        float hv = hr[k];
        a0 += hv * Wc2[k * 2 + 0];
        a1 += hv * Wc2[k * 2 + 1];
    }
    out[nrow * 2 + 0] = a0;
    out[nrow * 2 + 1] = a1;
}

// ---------------------------------------------------------------------------
extern "C" void kernel_launch(void* const* d_in, const int* in_sizes, int n_in,
                              void* d_out, int out_size, void* d_ws, size_t ws_size,
                              hipStream_t stream) {
    const float* x_user = (const float*)d_in[0];
    const float* x_pc   = (const float*)d_in[1];
    const float* x_url  = (const float*)d_in[2];
    const int* e_up_src = (const int*)d_in[3];
    const int* e_up_dst = (const int*)d_in[4];
    const int* e_uu_src = (const int*)d_in[5];
    const int* e_uu_dst = (const int*)d_in[6];
    const float* Wu = (const float*)d_in[7];   const float* bu = (const float*)d_in[8];
    const float* Wp = (const float*)d_in[9];   const float* bp = (const float*)d_in[10];
    const float* Wrf = (const float*)d_in[11]; const float* brf = (const float*)d_in[12];
    const float* Wl_pc = (const float*)d_in[13];  const float* bl_pc = (const float*)d_in[14];
    const float* Wr_pc = (const float*)d_in[15];
    const float* Wl_url = (const float*)d_in[16]; const float* bl_url = (const float*)d_in[17];
    const float* Wr_url = (const float*)d_in[18];
    const float* Wctx = (const float*)d_in[19];   const float* bctx = (const float*)d_in[20];
    const float* Wc1 = (const float*)d_in[21];    const float* bc1 = (const float*)d_in[22];
    const float* Wc2 = (const float*)d_in[23];    const float* bc2 = (const float*)d_in[24];
    float* out = (float*)d_out;

    // ---- workspace layout (floats) ----
    float* ws = (float*)d_ws;
    size_t off = 0;
    float* combined = ws + off; off += (size_t)NUSER * 384;  // [user | pc_ctx | url_ctx]
    float* mean_pc  = ws + off; off += (size_t)NPC * HD;     // also agg target
    float* cnt_pc   = ws + off; off += NPC;
    float* mean_url = ws + off; off += (size_t)NURL * HD;
    float* cnt_url  = ws + off; off += NURL;
    size_t zero_lo = (size_t)(mean_pc - ws);                 // mean/cnt block is contiguous
    size_t zero_n  = off - zero_lo;
    float* pc_a  = ws + off; off += (size_t)NPC * HD;
    float* pc_b  = ws + off; off += (size_t)NPC * HD;
    float* url_a = ws + off; off += (size_t)NURL * HD;
    float* url_b = ws + off; off += (size_t)NURL * HD;
    float* enr   = ws + off; off += (size_t)NUSER * HD;
    float* hbuf  = ws + off; off += (size_t)NUSER * 64;

    // zero atomic-accumulation targets (graph-capture-safe memset nodes)
    hipMemsetAsync(combined, 0, (size_t)NUSER * 384 * sizeof(float), stream);
    hipMemsetAsync(ws + zero_lo, 0, zero_n * sizeof(float), stream);

    // input projections (user writes cols 0..127 of combined, stride 384)
    hgs_proj_kernel<<<(NUSER * HD) / 256, 256, 0, stream>>>(x_user, Wu, bu, combined, NUSER, 6, 384);
    hgs_proj_kernel<<<(NPC * HD) / 256, 256, 0, stream>>>(x_pc, Wp, bp, pc_a, NPC, 4, HD);
    hgs_proj_kernel<<<(NURL * HD) / 256, 256, 0, stream>>>(x_url, Wrf, brf, url_a, NURL, 3, HD);

    // segment mean of user features (identical for both layers -> compute once)
    hgs_scatter_add_kernel<<<(EUP * 32) / 256, 256, 0, stream>>>(
        combined, 384, e_up_src, e_up_dst, mean_pc, HD, cnt_pc, EUP);
    hgs_scatter_add_kernel<<<(EUU * 32) / 256, 256, 0, stream>>>(
        combined, 384, e_uu_src, e_uu_dst, mean_url, HD, cnt_url, EUU);
    hgs_mean_div_kernel<<<(NPC * HD) / 256, 256, 0, stream>>>(mean_pc, cnt_pc, NPC);
    hgs_mean_div_kernel<<<(NURL * HD) / 256, 256, 0, stream>>>(mean_url, cnt_url, NURL);

    const int gpc  = (NPC + 31) / 32;
    const int gurl = (NURL + 31) / 32;
    // SAGE layer 0 (no residual): b = relu(mean@Wl0 + a@Wr0 + bl0)
    hgs_sage_fused_kernel<<<gpc, 256, 0, stream>>>(
        mean_pc, pc_a, Wl_pc, Wr_pc, bl_pc, pc_b, NPC, 0);
    hgs_sage_fused_kernel<<<gurl, 256, 0, stream>>>(
        mean_url, url_a, Wl_url, Wr_url, bl_url, url_b, NURL, 0);
    // SAGE layer 1 (residual): a = relu(mean@Wl1 + b@Wr1 + bl1 + b)
    hgs_sage_fused_kernel<<<gpc, 256, 0, stream>>>(
        mean_pc, pc_b, Wl_pc + HD * HD, Wr_pc + HD * HD, bl_pc + HD, pc_a, NPC, 1);
    hgs_sage_fused_kernel<<<gurl, 256, 0, stream>>>(
        mean_url, url_b, Wl_url + HD * HD, Wr_url + HD * HD, bl_url + HD, url_a, NURL, 1);

    // resource-context scatter into combined[:,128:256] and [:,256:384]
    hgs_scatter_add_kernel<<<(EUP * 32) / 256, 256, 0, stream>>>(
        pc_a, HD, e_up_dst, e_up_src, combined + HD, 384, nullptr, EUP);
    hgs_scatter_add_kernel<<<(EUU * 32) / 256, 256, 0, stream>>>(
        url_a, HD, e_uu_dst, e_uu_src, combined + 2 * HD, 384, nullptr, EUU);

    // enriched = combined @ Wctx + bctx   (K=384, 8 waves -> 128 cols)
    hgs_gemm_bias_kernel<<<NUSER / 32, 256, 32 * (384 + 4) * sizeof(float), stream>>>(
        combined, 384, 384, Wctx, bctx, enr, HD, NUSER, 0);
    // h = relu(enriched @ Wc1 + bc1)      (K=128, 4 waves -> 64 cols)
    hgs_gemm_bias_kernel<<<NUSER / 32, 128, 32 * (HD + 4) * sizeof(float), stream>>>(
        enr, HD, HD, Wc1, bc1, hbuf, 64, NUSER, 1);
    // out = h @ Wc2 + bc2
    hgs_head_kernel<<<(NUSER + 255) / 256, 256, 0, stream>>>(hbuf, Wc2, bc2, out, NUSER);
}